// LocalScoreMachine_49460843381564
// MI455X (gfx1250) — compile-verified
//
#include <hip/hip_runtime.h>

typedef __attribute__((ext_vector_type(2))) float v2f;
typedef __attribute__((ext_vector_type(8))) float v8f;

#define NWAVES 4
#define NB 8          // batch
#define NIMG 2048     // training images
#define HW 32
#define K27 27        // 3 channels * 3x3 patch
#define NEG_HUGE -3.402823466e38f

__global__ __launch_bounds__(128)
void lsm_score_kernel(const float* __restrict__ x,
                      const float* __restrict__ images,
                      const float* __restrict__ mu,
                      const float* __restrict__ sigma,
                      const int*   __restrict__ tptr,
                      float* __restrict__ out)
{
    __shared__ float xpatch[NB][28];           // zero-padded 3x3 patches of x, k = c*9+dh*3+dw
    __shared__ float sBSx[NB];                 // box3 of sum_c x^2 at this pixel
    __shared__ float red[NWAVES][16][NB][5];   // per-(wave,lane) partial {m,S,V0,V1,V2}

    const int p    = blockIdx.x;               // pixel id, 0..1023
    const int h    = p >> 5;
    const int w    = p & 31;
    const int tid  = threadIdx.x;
    const int wave = tid >> 5;
    const int lane = tid & 31;

    const int   t      = tptr[0];
    const float mu_t   = mu[t];
    const float sg     = sigma[t];
    const float sig2   = sg * sg;
    const float inv2s2 = 1.0f / (2.0f * sig2);
    const float mu2    = mu_t * mu_t;

    // ---- stage x patches (zero padding at image borders) ----
    if (tid < NB * 28) {
        int b = tid / 28, k = tid % 28;
        float v = 0.0f;
        if (k < K27) {
            int c = k / 9, r = k % 9;
            int hh = h + r / 3 - 1, ww = w + r % 3 - 1;
            if (hh >= 0 && hh < HW && ww >= 0 && ww < HW)
                v = x[((b * 3 + c) << 10) + (hh << 5) + ww];
        }
        xpatch[b][k] = v;
    }
    __syncthreads();
    if (tid < NB) {
        float s = 0.0f;
        #pragma unroll
        for (int k = 0; k < K27; ++k) s += xpatch[tid][k] * xpatch[tid][k];
        sBSx[tid] = s;
    }
    __syncthreads();

    // ---- A-matrix fragments (constant over all n-tiles) ----
    // A layout (16x4 f32): lanes 0-15 -> M=lane, K={0,1}; lanes 16-31 -> M=lane-16, K={2,3}
    const int mrow = lane & 15;
    const int koff = (lane < 16) ? 0 : 2;
    v2f afr[7];
    #pragma unroll
    for (int kc = 0; kc < 7; ++kc) {
        float a0 = (mrow < NB) ? xpatch[mrow][kc * 4 + koff + 0] : 0.0f;
        float a1 = (mrow < NB) ? xpatch[mrow][kc * 4 + koff + 1] : 0.0f;
        afr[kc] = (v2f){a0, a1};
    }
    float bsx[NB];
    #pragma unroll
    for (int b = 0; b < NB; ++b) bsx[b] = sBSx[b];

    // ---- online softmax state (per lane; lanes 0-15 carry real data) ----
    float mb[NB], Sb[NB], V0[NB], V1[NB], V2[NB];
    #pragma unroll
    for (int b = 0; b < NB; ++b) { mb[b] = NEG_HUGE; Sb[b] = 0.0f; V0[b] = V1[b] = V2[b] = 0.0f; }

    const int ncol = lane & 15;

    for (int tile = wave; tile < NIMG / 16; tile += NWAVES) {
        const int n = tile * 16 + ncol;
        const float* imgn = images + (((size_t)n * 3) << 10);

        // B-matrix fragments: B[k, ncol] = img patch element k (zero-padded)
        v2f bfr[7];
        float ssq = 0.0f;
        #pragma unroll
        for (int kc = 0; kc < 7; ++kc) {
            float e0 = 0.0f, e1 = 0.0f;
            #pragma unroll
            for (int s = 0; s < 2; ++s) {
                int k = kc * 4 + koff + s;
                float v = 0.0f;
                if (k < K27) {
                    int c = k / 9, r = k % 9;
                    int hh = h + r / 3 - 1, ww = w + r % 3 - 1;
                    if (hh >= 0 && hh < HW && ww >= 0 && ww < HW)
                        v = imgn[(c << 10) + (hh << 5) + ww];
                }
                if (s == 0) e0 = v; else e1 = v;
                ssq += v * v;
            }
            bfr[kc] = (v2f){e0, e1};
        }
        // full 27-element sum of squares: combine the two half-wave partials
        float bsi = ssq + __shfl_xor(ssq, 16);

        // center pixel values for the weighted mean
        float c0 = imgn[(h << 5) + w];
        float c1 = imgn[1024 + (h << 5) + w];
        float c2 = imgn[2048 + (h << 5) + w];

        // BD[b, n] via 7 chained fp32 WMMAs (K = 28, exact; pad K=27 is zero)
        v8f acc = {};
        #pragma unroll
        for (int kc = 0; kc < 7; ++kc)
            acc = __builtin_amdgcn_wmma_f32_16x16x4_f32(
                false, afr[kc], false, bfr[kc], (short)0, acc, false, false);

        // online softmax update (lane L<16 owns column n; rows b = acc[0..7])
        #pragma unroll
        for (int b = 0; b < NB; ++b) {
            float wfe  = -(bsx[b] - 2.0f * mu_t * acc[b] + mu2 * bsi) * inv2s2;
            float mn   = fmaxf(mb[b], wfe);
            float resc = __expf(mb[b] - mn);
            float wv   = __expf(wfe - mn);
            Sb[b] = Sb[b] * resc + wv;
            V0[b] = V0[b] * resc + wv * c0;
            V1[b] = V1[b] * resc + wv * c1;
            V2[b] = V2[b] * resc + wv * c2;
            mb[b] = mn;
        }
    }

    // ---- cross-lane / cross-wave log-sum-exp reduction ----
    if (lane < 16) {
        #pragma unroll
        for (int b = 0; b < NB; ++b) {
            red[wave][lane][b][0] = mb[b];
            red[wave][lane][b][1] = Sb[b];
            red[wave][lane][b][2] = V0[b];
            red[wave][lane][b][3] = V1[b];
            red[wave][lane][b][4] = V2[b];
        }
    }
    __syncthreads();

    if (tid < NB) {
        const int b = tid;
        float M = NEG_HUGE, S = 0.0f, A0 = 0.0f, A1 = 0.0f, A2 = 0.0f;
        for (int i = 0; i < NWAVES * 16; ++i) {
            int wv = i >> 4, ln = i & 15;
            float m2 = red[wv][ln][b][0];
            float Mn = fmaxf(M, m2);
            float s1 = __expf(M - Mn);
            float s2 = __expf(m2 - Mn);
            S  = S  * s1 + red[wv][ln][b][1] * s2;
            A0 = A0 * s1 + red[wv][ln][b][2] * s2;
            A1 = A1 * s1 + red[wv][ln][b][3] * s2;
            A2 = A2 * s1 + red[wv][ln][b][4] * s2;
            M = Mn;
        }
        const float invS   = 1.0f / S;
        const float invsg2 = 1.0f / sig2;
        // score = -(x - mu * V/S) / sig2   (max-shift cancels in the ratio)
        out[((b * 3 + 0) << 10) + p] = -(xpatch[b][4]  - mu_t * A0 * invS) * invsg2;
        out[((b * 3 + 1) << 10) + p] = -(xpatch[b][13] - mu_t * A1 * invS) * invsg2;
        out[((b * 3 + 2) << 10) + p] = -(xpatch[b][22] - mu_t * A2 * invS) * invsg2;
    }
}

extern "C" void kernel_launch(void* const* d_in, const int* in_sizes, int n_in,
                              void* d_out, int out_size, void* d_ws, size_t ws_size,
                              hipStream_t stream) {
    const float* x      = (const float*)d_in[0];
    const float* images = (const float*)d_in[1];
    const float* mu     = (const float*)d_in[2];
    const float* sigma  = (const float*)d_in[3];
    const int*   t      = (const int*)d_in[4];
    float* out = (float*)d_out;

    hipLaunchKernelGGL(lsm_score_kernel, dim3(1024), dim3(128), 0, stream,
                       x, images, mu, sigma, t, out);
}